// Sparsemax_23562190586290
// MI455X (gfx1250) — compile-verified
//
#include <hip/hip_runtime.h>
#include <stdint.h>

typedef __attribute__((ext_vector_type(2))) float v2f;
typedef __attribute__((ext_vector_type(4))) float v4f;
typedef __attribute__((ext_vector_type(8))) float v8f;

#define ROW_LEN  2048
#define WAVES    8          // waves per block (256 threads)
#define EPL      64         // elements per lane (2048 / 32)
#define NCHUNK   16         // 16-byte chunks per lane

__device__ __forceinline__ float wave_reduce_max(float v) {
    v = fmaxf(v, __shfl_xor(v, 1, 32));
    v = fmaxf(v, __shfl_xor(v, 2, 32));
    v = fmaxf(v, __shfl_xor(v, 4, 32));
    v = fmaxf(v, __shfl_xor(v, 8, 32));
    v = fmaxf(v, __shfl_xor(v, 16, 32));
    return v;
}

// Simultaneous wave-wide sum of (sp, cp) via one V_WMMA_F32_16X16X4_F32.
// A: A[m][0]=sp[m], A[m][1]=cp[m], A[m][2]=sp[m+16], A[m][3]=cp[m+16].
// B (VGPR0 = rows K0/K2, VGPR1 = rows K1/K3):
//   B[0]=B[2]=[n<8], B[1]=B[3]=[n>=8]
// => D[m][n<8] = sp[m]+sp[m+16], D[m][n>=8] = cp[m]+cp[m+16].
// Sum 8 D regs, fold lane-halves (xor 16), then xor 8 + select -> S,C on all lanes.
__device__ __forceinline__ void wave_dual_sum(float sp, float cp,
                                              float b0, float b1, bool lowcols,
                                              float& S, float& C) {
    v2f A; A[0] = sp; A[1] = cp;
    v2f B; B[0] = b0; B[1] = b1;
    v8f Z = {};
    v8f D = __builtin_amdgcn_wmma_f32_16x16x4_f32(
        /*neg_a=*/false, A, /*neg_b=*/false, B,
        /*c_mod=*/(short)0, Z, /*reuse_a=*/false, /*reuse_b=*/false);
    float t = ((D[0] + D[1]) + (D[2] + D[3])) + ((D[4] + D[5]) + (D[6] + D[7]));
    float u = t + __shfl_xor(t, 16, 32);
    float w = __shfl_xor(u, 8, 32);
    S = lowcols ? u : w;
    C = lowcols ? w : u;
}

__global__ __launch_bounds__(256)
void sparsemax_rows_kernel(const float* __restrict__ x,
                           float* __restrict__ out,
                           int nrows) {
    __shared__ float smem[WAVES * ROW_LEN];   // 64 KB

    const int tid  = threadIdx.x;
    const int wave = tid >> 5;
    const int lane = tid & 31;
    const int row  = blockIdx.x * WAVES + wave;
    if (row >= nrows) return;                 // whole-wave uniform exit

    const float* rowp  = x   + (size_t)row * ROW_LEN;
    float*       outp  = out + (size_t)row * ROW_LEN;
    float*       smrow = &smem[wave * ROW_LEN];

    // Constants for the dual-sum WMMA reduction
    const bool  lowcols = ((lane & 15) < 8);
    const float b0 = lowcols ? 1.0f : 0.0f;   // rows K0/K2: select cols 0..7 (sum lane)
    const float b1 = 1.0f - b0;               // rows K1/K3: select cols 8..15 (count lane)

    // ---------------- Async copy: global row -> LDS (ASYNCcnt path) -------
    // 16 chunks of 512B: lane L handles bytes [L*16, L*16+16) of each chunk.
    {
        unsigned ldsa = (unsigned)(uintptr_t)smrow + (unsigned)(lane * 16);
        unsigned long long ga = (unsigned long long)(uintptr_t)rowp
                              + (unsigned long long)(lane * 16);
        asm volatile(
            "global_load_async_to_lds_b128 %0, %1, off offset:0\n\t"
            "global_load_async_to_lds_b128 %0, %1, off offset:512\n\t"
            "global_load_async_to_lds_b128 %0, %1, off offset:1024\n\t"
            "global_load_async_to_lds_b128 %0, %1, off offset:1536\n\t"
            "global_load_async_to_lds_b128 %0, %1, off offset:2048\n\t"
            "global_load_async_to_lds_b128 %0, %1, off offset:2560\n\t"
            "global_load_async_to_lds_b128 %0, %1, off offset:3072\n\t"
            "global_load_async_to_lds_b128 %0, %1, off offset:3584\n\t"
            "global_load_async_to_lds_b128 %0, %1, off offset:4096\n\t"
            "global_load_async_to_lds_b128 %0, %1, off offset:4608\n\t"
            "global_load_async_to_lds_b128 %0, %1, off offset:5120\n\t"
            "global_load_async_to_lds_b128 %0, %1, off offset:5632\n\t"
            "global_load_async_to_lds_b128 %0, %1, off offset:6144\n\t"
            "global_load_async_to_lds_b128 %0, %1, off offset:6656\n\t"
            "global_load_async_to_lds_b128 %0, %1, off offset:7168\n\t"
            "global_load_async_to_lds_b128 %0, %1, off offset:7680"
            :
            : "v"(ldsa), "v"(ga)
            : "memory");
        asm volatile("s_wait_asynccnt 0" ::: "memory");
    }

    // ---------------- LDS -> registers (64 floats per lane) ---------------
    float v[EPL];
#pragma unroll
    for (int c = 0; c < NCHUNK; ++c) {
        v4f q = *(const v4f*)(smrow + c * 128 + lane * 4);
        v[c * 4 + 0] = q.x; v[c * 4 + 1] = q.y;
        v[c * 4 + 2] = q.z; v[c * 4 + 3] = q.w;
    }

    // ---------------- Row max, shift, per-lane partial sum ----------------
    float m = v[0];
#pragma unroll
    for (int i = 1; i < EPL; ++i) m = fmaxf(m, v[i]);
    m = wave_reduce_max(m);

    float ps = 0.0f;
#pragma unroll
    for (int i = 0; i < EPL; ++i) { v[i] -= m; ps += v[i]; }

    // Wave-wide total sum (dual helper with cp = 0)
    float total, dummy;
    wave_dual_sum(ps, 0.0f, b0, b1, lowcols, total, dummy);

    // ---------------- Michelot iteration for the sparsemax threshold ------
    float tau  = (total - 1.0f) * (1.0f / (float)ROW_LEN);
    int cprev  = ROW_LEN;
#pragma unroll 1
    for (int it = 0; it < 64; ++it) {
        float s = 0.0f, cnt = 0.0f;
#pragma unroll
        for (int i = 0; i < EPL; ++i) {
            bool g = v[i] > tau;          // select, no divergence
            s   += g ? v[i] : 0.0f;
            cnt += g ? 1.0f : 0.0f;       // count <= 2048, exact in f32
        }
        float S, C;
        wave_dual_sum(s, cnt, b0, b1, lowcols, S, C);  // one WMMA, 2 shuffles
        tau = (S - 1.0f) / C;
        int ci = (int)C;                  // wave-uniform
        if (ci == cprev) break;           // active set unchanged -> converged
        cprev = ci;
    }

    // ---------------- Output: max(0, x - tau), streaming NT stores --------
#pragma unroll
    for (int c = 0; c < NCHUNK; ++c) {
        v4f r;
        r.x = fmaxf(v[c * 4 + 0] - tau, 0.0f);
        r.y = fmaxf(v[c * 4 + 1] - tau, 0.0f);
        r.z = fmaxf(v[c * 4 + 2] - tau, 0.0f);
        r.w = fmaxf(v[c * 4 + 3] - tau, 0.0f);
        __builtin_nontemporal_store(r, (v4f*)(outp + c * 128 + lane * 4));
    }
}

extern "C" void kernel_launch(void* const* d_in, const int* in_sizes, int n_in,
                              void* d_out, int out_size, void* d_ws, size_t ws_size,
                              hipStream_t stream) {
    (void)n_in; (void)out_size; (void)d_ws; (void)ws_size;
    const float* x = (const float*)d_in[0];
    float* out     = (float*)d_out;
    int nrows  = in_sizes[0] / ROW_LEN;            // 65536
    int blocks = (nrows + WAVES - 1) / WAVES;      // 8192
    sparsemax_rows_kernel<<<blocks, 256, 0, stream>>>(x, out, nrows);
}